// msgPassLayer_82231443849482
// MI455X (gfx1250) — compile-verified
//
#include <hip/hip_runtime.h>

typedef __attribute__((ext_vector_type(2))) float v2f;
typedef __attribute__((ext_vector_type(8))) float v8f;

#define NSITES_C 262144
#define ZNBR 12
#define NPAIR 6
#define WAVES_PER_BLOCK 8
#define THREADS_PER_BLOCK 256

// numerically-stable softplus in base-2, using RAW v_exp_f32/v_log_f32:
//   softplus(x) = max(x,0) + ln2 * log2(1 + exp2(-|x| * log2e))
// log2 argument is in [1,2] -> no denorm guard needed (unlike __logf's lowering)
__device__ __forceinline__ float softplus_f(float x) {
    const float LOG2E = 1.4426950408889634f;
    const float LN2   = 0.6931471805599453f;
    float ax = __builtin_fabsf(x);
    float e  = __builtin_amdgcn_exp2f(-ax * LOG2E);   // bare v_exp_f32
    float lg = __builtin_amdgcn_logf(1.0f + e);       // bare v_log_f32
    return __builtin_fmaf(LN2, lg, fmaxf(x, 0.0f));
}

// ---- one-time reduction of the tiny constants into workspace:
// ws[0..127]   = Weff[o,k] = sum_j Weights[j,o,k]   (8x16)
// ws[128..135] = beff[o]   = sum_j bias[j,o]        (8)
__global__ void weff_setup_kernel(const float* __restrict__ W,
                                  const float* __restrict__ bias,
                                  float* __restrict__ ws)
{
    const int t = threadIdx.x;
    if (t < 128) {
        float s = 0.0f;
        #pragma unroll
        for (int j = 0; j < 8; ++j) s += W[j * 128 + t];
        ws[t] = s;
    } else if (t < 136) {
        const int o = t - 128;
        float s = 0.0f;
        #pragma unroll
        for (int j = 0; j < 8; ++j) s += bias[j * 8 + o];
        ws[t] = s;
    }
}

__global__ __launch_bounds__(THREADS_PER_BLOCK)
void msgpass_wmma_kernel(const float* __restrict__ In,     // (4, 8, NSITES)
                         const int*   __restrict__ NN,     // (13, NSITES)
                         const float* __restrict__ WS,     // [Weff 8x16 | beff 8]
                         float*       __restrict__ out)    // (4, 8, NSITES)
{
    const int lane = threadIdx.x & 31;
    const int wid  = threadIdx.x >> 5;
    const int task = blockIdx.x * WAVES_PER_BLOCK + wid;  // exact fit: no tail
    const int b    = task & 3;
    const int tile = task >> 2;
    const int s0   = tile << 4;       // first site of this 16-site tile
    const int m    = lane & 15;       // site-in-tile (B/C/D lanes); row idx for A lanes
    const int half = lane >> 4;       // A/B K-half selector within a 16x4 chunk
    const int o    = m & 7;           // output channel this A-lane's row maps to

    const float* Inb = In + (size_t)b * 8 * NSITES_C;

    // ---- per-lane A-matrix slices from precomputed Weff.
    // A 16x4 f32 layout: lanes 0-15 hold K=v, lanes 16-31 hold K=v+2.
    float wself[2][2];   // Ws = Weff[:, 0:8],  chunk ch: k = ch*4 + v + 2*half
    float wn[2][2];      // Wn = Weff[:, 8:16], same local k
    #pragma unroll
    for (int ch = 0; ch < 2; ++ch) {
        #pragma unroll
        for (int v = 0; v < 2; ++v) {
            const int k = ch * 4 + v + 2 * half;
            wself[ch][v] = WS[o * 16 + k];
            wn[ch][v]    = WS[o * 16 + 8 + k];
        }
    }

    // Block-diagonal A for a z-pair (branchless select keeps EXEC all-ones):
    //   even z: rows 0-7 <- Wn (K 0-7),  rows 8-15 <- 0
    //   odd  z: rows 0-7 <- 0,           rows 8-15 <- Wn (K 8-15)
    const bool lowrow = (m < 8);
    float wAe[2][2], wAo[2][2];
    #pragma unroll
    for (int ch = 0; ch < 2; ++ch)
        #pragma unroll
        for (int v = 0; v < 2; ++v) {
            wAe[ch][v] = lowrow ? wn[ch][v] : 0.0f;
            wAo[ch][v] = lowrow ? 0.0f     : wn[ch][v];
        }

    // ---- self part, with beff folded in as the WMMA C operand:
    // dself[k] starts as beff[k] (uniform across lanes -> hits rows k AND k+8,
    // i.e. both the even-z and odd-z lane halves), then accumulates Ws x self.
    // Ws is DUPLICATED on A rows 0-7 / 8-15 so dself mirrors across halves.
    v8f dself;
    #pragma unroll
    for (int k = 0; k < 8; ++k) dself[k] = WS[128 + k];   // beff (scalar loads)

    #pragma unroll
    for (int ch = 0; ch < 2; ++ch) {
        v2f a = { wself[ch][0], wself[ch][1] };
        const int i0 = ch * 4 + 0 + 2 * half;
        const int i1 = ch * 4 + 1 + 2 * half;
        v2f bb = { Inb[(size_t)i0 * NSITES_C + s0 + m],
                   Inb[(size_t)i1 * NSITES_C + s0 + m] };
        dself = __builtin_amdgcn_wmma_f32_16x16x4_f32(
                    false, a, false, bb, (short)0, dself, false, false);
    }

    // ---- neighbor indices for all 12 z (coalesced; lanes 16-31 mirror 0-15)
    int idxv[ZNBR];
    #pragma unroll
    for (int z = 0; z < ZNBR; ++z)
        idxv[z] = NN[(size_t)(z + 1) * NSITES_C + s0 + m];

    float acc[8];
    #pragma unroll
    for (int k = 0; k < 8; ++k) acc[k] = 0.0f;

    // ---- 6 z-pairs: 4 WMMAs each (K=16 block-diagonal), softplus on ALL 32 lanes
    #pragma unroll 2
    for (int p = 0; p < NPAIR; ++p) {
        const int idxE = idxv[2 * p];
        const int idxO = idxv[2 * p + 1];

        // prefetch next pair's gather lines into L0 while this pair computes
        if (p + 1 < NPAIR) {
            const int ie = idxv[2 * p + 2];
            const int io = idxv[2 * p + 3];
            __builtin_prefetch(&Inb[(size_t)(0 + 2 * half) * NSITES_C + ie], 0, 0);
            __builtin_prefetch(&Inb[(size_t)(4 + 2 * half) * NSITES_C + ie], 0, 0);
            __builtin_prefetch(&Inb[(size_t)(0 + 2 * half) * NSITES_C + io], 0, 0);
            __builtin_prefetch(&Inb[(size_t)(4 + 2 * half) * NSITES_C + io], 0, 0);
        }

        v8f d = dself;  // C = self part + beff (mirrored across halves)
        // even z -> D rows 0-7 (lanes 0-15)
        #pragma unroll
        for (int ch = 0; ch < 2; ++ch) {
            v2f a = { wAe[ch][0], wAe[ch][1] };
            const int i0 = ch * 4 + 0 + 2 * half;
            const int i1 = ch * 4 + 1 + 2 * half;
            v2f bb = { Inb[(size_t)i0 * NSITES_C + idxE],
                       Inb[(size_t)i1 * NSITES_C + idxE] };
            d = __builtin_amdgcn_wmma_f32_16x16x4_f32(
                    false, a, false, bb, (short)0, d, false, false);
        }
        // odd z -> D rows 8-15 (lanes 16-31)
        #pragma unroll
        for (int ch = 0; ch < 2; ++ch) {
            v2f a = { wAo[ch][0], wAo[ch][1] };
            const int i0 = ch * 4 + 0 + 2 * half;
            const int i1 = ch * 4 + 1 + 2 * half;
            v2f bb = { Inb[(size_t)i0 * NSITES_C + idxO],
                       Inb[(size_t)i1 * NSITES_C + idxO] };
            d = __builtin_amdgcn_wmma_f32_16x16x4_f32(
                    false, a, false, bb, (short)0, d, false, false);
        }

        // softplus + accumulate: lanes 0-15 sum even z's, lanes 16-31 odd z's
        // (bias already inside d via the C-operand fold)
        #pragma unroll
        for (int k = 0; k < 8; ++k)
            acc[k] += softplus_f(d[k]);
    }

    // combine lane halves: acc(lanes 0-15) += acc(lanes 16-31)  [even+odd z sums]
    #pragma unroll
    for (int k = 0; k < 8; ++k)
        acc[k] += __shfl_xor(acc[k], 16, 32);

    // store: VGPR k, lanes 0-15 = (o=k, site=s0+m)
    if (lane < 16) {
        float* ob = out + (size_t)b * 8 * NSITES_C + s0 + m;
        #pragma unroll
        for (int k = 0; k < 8; ++k) ob[(size_t)k * NSITES_C] = acc[k];
    }
}

extern "C" void kernel_launch(void* const* d_in, const int* in_sizes, int n_in,
                              void* d_out, int out_size, void* d_ws, size_t ws_size,
                              hipStream_t stream) {
    const float* In   = (const float*)d_in[0];
    const int*   NN   = (const int*)d_in[1];
    const float* W    = (const float*)d_in[2];
    const float* bias = (const float*)d_in[3];
    float*       out  = (float*)d_out;
    float*       ws   = (float*)d_ws;   // 136 floats: Weff(128) | beff(8)

    // one-time constant reduction (same stream -> ordered before main kernel)
    weff_setup_kernel<<<1, 160, 0, stream>>>(W, bias, ws);

    const int tasks  = 4 * (NSITES_C / 16);          // 65536 waves
    const int blocks = tasks / WAVES_PER_BLOCK;      // 8192 blocks of 256 (8 waves)
    msgpass_wmma_kernel<<<blocks, THREADS_PER_BLOCK, 0, stream>>>(In, NN, ws, out);
}